// MACELayer_28879360098431
// MI455X (gfx1250) — compile-verified
//
#include <hip/hip_runtime.h>

// MACE layer for gfx1250 (MI455X): fp32 WMMA 16x16x4, pair-packed B operands
// (1x global_load_b64 per WMMA), K-outer multi-accumulator GEMM loops,
// fused edge pipeline with L2-resident global_atomic_add_f32 scatter.

#define N_NODES 20000
#define E_EDGES 320000
#define INV_SQRT3 0.57735026918962576451f

typedef __attribute__((ext_vector_type(2))) float v2f;
typedef __attribute__((ext_vector_type(8))) float v8f;

static __device__ __forceinline__ v8f wmma4(v2f a, v2f b, v8f c) {
  // D = A(16x4 f32) * B(4x16 f32) + C(16x16 f32)
  return __builtin_amdgcn_wmma_f32_16x16x4_f32(
      false, a, false, b, (short)0, c, false, false);
}

// B fragment from pair-packed weights: WP[k/2][col][2] = {W[k][col], W[k+1][col]}
// k0 is always even (k0 = kk + 2*kh, kk % 4 == 0) -> single aligned b64 load.
static __device__ __forceinline__ v2f ldb(const float* __restrict__ WP,
                                          int ncol, int k0, int col) {
  return *(const v2f*)(WP + (size_t)(k0 >> 1) * (2 * ncol) + col * 2);
}

static __device__ __forceinline__ v2f ld2(const float* __restrict__ p) {
  v2f r; r.x = p[0]; r.y = p[1]; return r;
}

// ---------------------------------------------------------------------------
// Prep: pack W (K x ncol, row-major) into k-pair layout for b64 B fragments.
// ---------------------------------------------------------------------------
__global__ void k_pack_pairs(const float* __restrict__ W, float* __restrict__ WP,
                             int K, int ncol) {
  const int i = blockIdx.x * blockDim.x + threadIdx.x;
  if (i >= K * ncol) return;
  const int k = i / ncol, c = i - k * ncol;
  WP[(size_t)(k >> 1) * (2 * ncol) + c * 2 + (k & 1)] = W[i];
}

// ---------------------------------------------------------------------------
// Kernel A: x0 = x0_in @ W_pre0 ; x1[:,d,m] = sum_c x1_in[:,c,m] W_pre1[c,d]
// Also zeroes t0/t1. One wave per 16-node tile, 4 waves / block.
// ---------------------------------------------------------------------------
__global__ __launch_bounds__(128) void k_node_pre(
    const float* __restrict__ nf, const float* __restrict__ Wp0P,
    const float* __restrict__ Wp1P, float* __restrict__ x0,
    float* __restrict__ x1, float* __restrict__ t0, float* __restrict__ t1) {
  const int wave = threadIdx.x >> 5;
  const int lane = threadIdx.x & 31;
  const int tile = blockIdx.x * 4 + wave;
  if (tile >= (N_NODES / 16)) return;
  const int nbase = tile * 16;
  const int lrow  = lane & 15;
  const int kh    = lane >> 4;
  const int arow  = nbase + lrow;

  // x0: K-outer, 4 column-tile accumulators
  {
    v8f acc[4] = {{}, {}, {}, {}};
    for (int kk = 0; kk < 64; kk += 4) {
      const int k0 = kk + 2 * kh;
      const v2f a = ld2(&nf[(size_t)arow * 256 + k0]);
#pragma unroll
      for (int j = 0; j < 4; ++j)
        acc[j] = wmma4(a, ldb(Wp0P, 64, k0, 16 * j + lrow), acc[j]);
    }
#pragma unroll
    for (int j = 0; j < 4; ++j)
#pragma unroll
      for (int v = 0; v < 8; ++v)
        x0[(size_t)(nbase + v + 8 * kh) * 64 + 16 * j + lrow] = acc[j][v];
  }

  // x1: per Cartesian component m (stride-3 A loads)
  for (int m = 0; m < 3; ++m) {
    v8f acc[4] = {{}, {}, {}, {}};
    for (int kk = 0; kk < 64; kk += 4) {
      const int k0 = kk + 2 * kh;
      v2f a;
      a.x = nf[(size_t)arow * 256 + 64 + k0 * 3 + m];
      a.y = nf[(size_t)arow * 256 + 64 + (k0 + 1) * 3 + m];
#pragma unroll
      for (int j = 0; j < 4; ++j)
        acc[j] = wmma4(a, ldb(Wp1P, 64, k0, 16 * j + lrow), acc[j]);
    }
#pragma unroll
    for (int j = 0; j < 4; ++j)
#pragma unroll
      for (int v = 0; v < 8; ++v)
        x1[(size_t)(nbase + v + 8 * kh) * 192 + (16 * j + lrow) * 3 + m] = acc[j][v];
  }

  // zero scatter accumulators (float4 stores)
  float4* t0v = (float4*)(t0 + (size_t)nbase * 64);
  float4* t1v = (float4*)(t1 + (size_t)nbase * 192);
  const float4 z4 = make_float4(0.f, 0.f, 0.f, 0.f);
  for (int i = lane; i < 16 * 64 / 4; i += 32)  t0v[i] = z4;
  for (int i = lane; i < 16 * 192 / 4; i += 32) t1v[i] = z4;
}

// ---------------------------------------------------------------------------
// Kernel B: fused edge pipeline.
//   w = silu(rb @ W1) @ W2  (WMMA, all 16 w tiles in registers),
//   messages + scatter-add into t0/t1 (L2-resident atomics).
// One wave per 16-edge tile, 4 waves / block.
// ---------------------------------------------------------------------------
__global__ __launch_bounds__(128) void k_edge(
    const float* __restrict__ sph, const float* __restrict__ rb,
    const int* __restrict__ snd, const int* __restrict__ rcv,
    const float* __restrict__ W1P, const float* __restrict__ W2P,
    const float* __restrict__ x0, const float* __restrict__ x1,
    float* __restrict__ t0, float* __restrict__ t1) {
  __shared__ float hidden[4][16][64];
  const int wave  = threadIdx.x >> 5;
  const int lane  = threadIdx.x & 31;
  const int lrow  = lane & 15;
  const int kh    = lane >> 4;
  const int ebase = (blockIdx.x * 4 + wave) * 16;

  // GEMM1 (16x8 @ 8x64) + silu -> LDS
  {
    v8f acc[4] = {{}, {}, {}, {}};
#pragma unroll
    for (int kk = 0; kk < 8; kk += 4) {
      const int k0 = kk + 2 * kh;
      const v2f a = ld2(&rb[(size_t)(ebase + lrow) * 8 + k0]);
#pragma unroll
      for (int j = 0; j < 4; ++j)
        acc[j] = wmma4(a, ldb(W1P, 64, k0, 16 * j + lrow), acc[j]);
    }
#pragma unroll
    for (int j = 0; j < 4; ++j)
#pragma unroll
      for (int v = 0; v < 8; ++v) {
        float h = acc[j][v];
        h = h / (1.f + __expf(-h));  // silu
        hidden[wave][v + 8 * kh][16 * j + lrow] = h;
      }
  }
  __syncthreads();

  // GEMM2 (16x64 @ 64x256): K-outer, one ds_load_b64 A + 16 b64 B per 16 WMMA.
  v8f w[16] = {{}, {}, {}, {}, {}, {}, {}, {}, {}, {}, {}, {}, {}, {}, {}, {}};
  for (int kk = 0; kk < 64; kk += 4) {
    const int k0 = kk + 2 * kh;
    const v2f a = ld2(&hidden[wave][lrow][k0]);
#pragma unroll
    for (int j = 0; j < 16; ++j)
      w[j] = wmma4(a, ldb(W2P, 256, k0, 16 * j + lrow), w[j]);
  }

  // Messages + scatter. Lane holds channel c = 16j+lrow for 8 edges (v+8*kh);
  // the four w-blocks for channel c live in tiles j, j+4, j+8, j+12.
#pragma unroll
  for (int v = 0; v < 8; ++v) {
    const int e = ebase + v + 8 * kh;
    const int s = snd[e];
    const int r = rcv[e];
    const float Y0  = sph[(size_t)e * 4 + 0];
    const float Y1x = sph[(size_t)e * 4 + 1];
    const float Y1y = sph[(size_t)e * 4 + 2];
    const float Y1z = sph[(size_t)e * 4 + 3];
#pragma unroll
    for (int j = 0; j < 4; ++j) {
      const int c = 16 * j + lrow;
      const float w0 = w[j][v], w1 = w[j + 4][v];
      const float w2 = w[j + 8][v], w3 = w[j + 12][v];
      const float h0  = x0[(size_t)s * 64 + c];
      const float h1x = x1[(size_t)s * 192 + c * 3 + 0];
      const float h1y = x1[(size_t)s * 192 + c * 3 + 1];
      const float h1z = x1[(size_t)s * 192 + c * 3 + 2];
      const float dot = h1x * Y1x + h1y * Y1y + h1z * Y1z;
      atomicAdd(&t0[(size_t)r * 64 + c], w0 * h0 * Y0 + w1 * dot * INV_SQRT3);
      const size_t b1 = (size_t)r * 192 + c * 3;
      atomicAdd(&t1[b1 + 0], w2 * h0 * Y1x + w3 * h1x * Y0);
      atomicAdd(&t1[b1 + 1], w2 * h0 * Y1y + w3 * h1y * Y0);
      atomicAdd(&t1[b1 + 2], w2 * h0 * Y1z + w3 * h1z * Y0);
    }
  }
}

// ---------------------------------------------------------------------------
// Kernel C: node post-processing. a-GEMMs -> polynomial gating (D-layout
// registers) -> Wr K=640 WMMA chains with on-the-fly outer-product A (built
// once per k-step, shared by 4 column tiles) -> LDS transpose -> f GEMMs.
// One wave per 16-node tile, 2 waves / block.
// ---------------------------------------------------------------------------
__global__ __launch_bounds__(64) void k_node_post(
    const float* __restrict__ nf, const float* __restrict__ attrs,
    const float* __restrict__ Wpost0P, const float* __restrict__ Wpost1P,
    const float* __restrict__ Wc0, const float* __restrict__ Wc1,
    const float* __restrict__ Wr0P, const float* __restrict__ Wr1P,
    const float* __restrict__ Wf0P, const float* __restrict__ Wf1P,
    const float* __restrict__ t0, const float* __restrict__ t1,
    float* __restrict__ out) {
  __shared__ float s0s[2][16][64];
  __shared__ float s1s[2][16][192];
  const int wave  = threadIdx.x >> 5;
  const int lane  = threadIdx.x & 31;
  const int lrow  = lane & 15;
  const int kh    = lane >> 4;
  const int nbase = (blockIdx.x * 2 + wave) * 16;
  const int arow  = nbase + lrow;

  // Phase 1: a0 = (t0/16) @ Wpost0, a1[m] = (t1[...,m]/16) @ Wpost1
  v8f a0[4]    = {{}, {}, {}, {}};
  v8f a1[3][4] = {{{}, {}, {}, {}}, {{}, {}, {}, {}}, {{}, {}, {}, {}}};
  for (int kk = 0; kk < 64; kk += 4) {
    const int k0 = kk + 2 * kh;
    const v2f at0 = ld2(&t0[(size_t)arow * 64 + k0]);
#pragma unroll
    for (int j = 0; j < 4; ++j)
      a0[j] = wmma4(at0, ldb(Wpost0P, 64, k0, 16 * j + lrow), a0[j]);
#pragma unroll
    for (int m = 0; m < 3; ++m) {
      v2f at1;
      at1.x = t1[(size_t)arow * 192 + k0 * 3 + m];
      at1.y = t1[(size_t)arow * 192 + (k0 + 1) * 3 + m];
#pragma unroll
      for (int j = 0; j < 4; ++j)
        a1[m][j] = wmma4(at1, ldb(Wpost1P, 64, k0, 16 * j + lrow), a1[m][j]);
    }
  }

  // Phase 2: pointwise polynomial gating; overwrite a0/a1 with out0/out1.
#pragma unroll
  for (int j = 0; j < 4; ++j) {
    const int col = 16 * j + lrow;
#pragma unroll
    for (int v = 0; v < 8; ++v) {
      const int node = nbase + v + 8 * kh;
      const float s  = a0[j][v] * (1.f / 16.f);
      const float ax = a1[0][j][v] * (1.f / 16.f);
      const float ay = a1[1][j][v] * (1.f / 16.f);
      const float az = a1[2][j][v] * (1.f / 16.f);
      const float q  = ax * ax + ay * ay + az * az;
      float c0 = 0.f, c1 = 0.f, c2 = 0.f, d0 = 0.f, d1 = 0.f, d2 = 0.f;
#pragma unroll
      for (int z = 0; z < 10; ++z) {
        const float t = attrs[(size_t)node * 10 + z];
        c0 += t * Wc0[z * 64 + col];
        c1 += t * Wc0[640 + z * 64 + col];
        c2 += t * Wc0[1280 + z * 64 + col];
        d0 += t * Wc1[z * 64 + col];
        d1 += t * Wc1[640 + z * 64 + col];
        d2 += t * Wc1[1280 + z * 64 + col];
      }
      const float s2 = s * s;
      const float g  = d2 * (s2 + q) + d1 * s + d0;
      a0[j][v]    = c2 * (s2 * s + s * q) + c1 * (s2 + q) + c0 * s;
      a1[0][j][v] = g * ax;
      a1[1][j][v] = g * ay;
      a1[2][j][v] = g * az;
    }
  }

  // Phase 3: Wr chains, K=640. A = x (outer) attrs built once per k-step.
  for (int kk = 0; kk < 640; kk += 4) {
    const int k0 = kk + 2 * kh;
    const int u0 = k0 / 10, z0 = k0 - u0 * 10;
    const int k1 = k0 + 1;
    const int u1 = k1 / 10, z1 = k1 - u1 * 10;
    v2f a;
    a.x = nf[(size_t)arow * 256 + u0] * attrs[(size_t)arow * 10 + z0];
    a.y = nf[(size_t)arow * 256 + u1] * attrs[(size_t)arow * 10 + z1];
#pragma unroll
    for (int j = 0; j < 4; ++j)
      a0[j] = wmma4(a, ldb(Wr0P, 64, k0, 16 * j + lrow), a0[j]);
  }
#pragma unroll
  for (int m = 0; m < 3; ++m) {
    for (int kk = 0; kk < 640; kk += 4) {
      const int k0 = kk + 2 * kh;
      const int u0 = k0 / 10, z0 = k0 - u0 * 10;
      const int k1 = k0 + 1;
      const int u1 = k1 / 10, z1 = k1 - u1 * 10;
      v2f a;
      a.x = nf[(size_t)arow * 256 + 64 + u0 * 3 + m] * attrs[(size_t)arow * 10 + z0];
      a.y = nf[(size_t)arow * 256 + 64 + u1 * 3 + m] * attrs[(size_t)arow * 10 + z1];
#pragma unroll
      for (int j = 0; j < 4; ++j)
        a1[m][j] = wmma4(a, ldb(Wr1P, 64, k0, 16 * j + lrow), a1[m][j]);
    }
  }

  // Stage for D-layout -> A-layout transpose
#pragma unroll
  for (int j = 0; j < 4; ++j) {
    const int col = 16 * j + lrow;
#pragma unroll
    for (int v = 0; v < 8; ++v) {
      const int M = v + 8 * kh;
      s0s[wave][M][col] = a0[j][v];
      s1s[wave][M][col * 3 + 0] = a1[0][j][v];
      s1s[wave][M][col * 3 + 1] = a1[1][j][v];
      s1s[wave][M][col * 3 + 2] = a1[2][j][v];
    }
  }
  __syncthreads();

  // Phase 4: f0 = out0 @ Wf0 ; f1[:,d,m] = sum_c out1[:,c,m] Wf1[c,d]
  {
    v8f acc[4] = {{}, {}, {}, {}};
    for (int kk = 0; kk < 64; kk += 4) {
      const int k0 = kk + 2 * kh;
      const v2f a = ld2(&s0s[wave][lrow][k0]);
#pragma unroll
      for (int j = 0; j < 4; ++j)
        acc[j] = wmma4(a, ldb(Wf0P, 64, k0, 16 * j + lrow), acc[j]);
    }
#pragma unroll
    for (int j = 0; j < 4; ++j)
#pragma unroll
      for (int v = 0; v < 8; ++v)
        out[(size_t)(nbase + v + 8 * kh) * 256 + 16 * j + lrow] = acc[j][v];
  }
#pragma unroll
  for (int m = 0; m < 3; ++m) {
    v8f acc[4] = {{}, {}, {}, {}};
    for (int kk = 0; kk < 64; kk += 4) {
      const int k0 = kk + 2 * kh;
      v2f a;
      a.x = s1s[wave][lrow][k0 * 3 + m];
      a.y = s1s[wave][lrow][(k0 + 1) * 3 + m];
#pragma unroll
      for (int j = 0; j < 4; ++j)
        acc[j] = wmma4(a, ldb(Wf1P, 64, k0, 16 * j + lrow), acc[j]);
    }
#pragma unroll
    for (int j = 0; j < 4; ++j)
#pragma unroll
      for (int v = 0; v < 8; ++v)
        out[(size_t)(nbase + v + 8 * kh) * 256 + 64 + (16 * j + lrow) * 3 + m] =
            acc[j][v];
  }
}

extern "C" void kernel_launch(void* const* d_in, const int* in_sizes, int n_in,
                              void* d_out, int out_size, void* d_ws, size_t ws_size,
                              hipStream_t stream) {
  (void)in_sizes; (void)n_in; (void)out_size; (void)ws_size;
  const float* nf     = (const float*)d_in[0];
  const float* attrs  = (const float*)d_in[1];
  const float* sph    = (const float*)d_in[2];
  const float* rb     = (const float*)d_in[3];
  const int*   snd    = (const int*)d_in[4];
  const int*   rcv    = (const int*)d_in[5];
  const float* Wpre0  = (const float*)d_in[6];
  const float* Wpre1  = (const float*)d_in[7];
  const float* W1     = (const float*)d_in[8];
  const float* W2     = (const float*)d_in[9];
  const float* Wpost0 = (const float*)d_in[10];
  const float* Wpost1 = (const float*)d_in[11];
  const float* Wc0    = (const float*)d_in[12];
  const float* Wc1    = (const float*)d_in[13];
  const float* Wr0    = (const float*)d_in[14];
  const float* Wr1    = (const float*)d_in[15];
  const float* Wf0    = (const float*)d_in[16];
  const float* Wf1    = (const float*)d_in[17];

  float* ws = (float*)d_ws;
  size_t off = 0;
  float* x0 = ws + off; off += (size_t)N_NODES * 64;    // 1.28M
  float* x1 = ws + off; off += (size_t)N_NODES * 192;   // 3.84M
  float* t0 = ws + off; off += (size_t)N_NODES * 64;
  float* t1 = ws + off; off += (size_t)N_NODES * 192;
  float* Wpre0P  = ws + off; off += 64 * 64;
  float* Wpre1P  = ws + off; off += 64 * 64;
  float* W1P     = ws + off; off += 8 * 64;
  float* W2P     = ws + off; off += 64 * 256;
  float* Wpost0P = ws + off; off += 64 * 64;
  float* Wpost1P = ws + off; off += 64 * 64;
  float* Wr0P    = ws + off; off += 640 * 64;
  float* Wr1P    = ws + off; off += 640 * 64;
  float* Wf0P    = ws + off; off += 64 * 64;
  float* Wf1P    = ws + off; off += 64 * 64;

  // Pack weights into k-pair layout (B fragments become single b64 loads).
  auto pack = [&](const float* W, float* WP, int K, int ncol) {
    const int n = K * ncol;
    k_pack_pairs<<<(n + 255) / 256, 256, 0, stream>>>(W, WP, K, ncol);
  };
  pack(Wpre0, Wpre0P, 64, 64);
  pack(Wpre1, Wpre1P, 64, 64);
  pack(W1, W1P, 8, 64);
  pack(W2, W2P, 64, 256);
  pack(Wpost0, Wpost0P, 64, 64);
  pack(Wpost1, Wpost1P, 64, 64);
  pack(Wr0, Wr0P, 640, 64);
  pack(Wr1, Wr1P, 640, 64);
  pack(Wf0, Wf0P, 64, 64);
  pack(Wf1, Wf1P, 64, 64);

  const int ntiles = N_NODES / 16;  // 1250
  k_node_pre<<<(ntiles + 3) / 4, 128, 0, stream>>>(nf, Wpre0P, Wpre1P, x0, x1, t0, t1);
  k_edge<<<E_EDGES / 64, 128, 0, stream>>>(sph, rb, snd, rcv, W1P, W2P, x0, x1, t0, t1);
  k_node_post<<<ntiles / 2, 64, 0, stream>>>(nf, attrs, Wpost0P, Wpost1P, Wc0, Wc1,
                                             Wr0P, Wr1P, Wf0P, Wf1P, t0, t1,
                                             (float*)d_out);
}